// HGDCODE_32933809225769
// MI455X (gfx1250) — compile-verified
//
#include <hip/hip_runtime.h>

// ---------------- problem constants ----------------
#define B_    8
#define T_    64
#define N_    1024
#define DIN_  16
#define DR_   64

typedef __attribute__((ext_vector_type(2))) float v2f;
typedef __attribute__((ext_vector_type(8))) float v8f;

// =====================================================================
// Generic batched fp32 WMMA GEMM:  C[bz] = act(A[bz] @ B[bz] + bias) (+ addsrc)
//   grid.x = Ncols/16, grid.y = M/16, grid.z = batch ; block = 32 (1 wave)
//   act: 0=none, 1=tanh, 2=sigmoid.  addsrc added AFTER activation (used with
//   act==0 only, for the Euler residual  h + f(h)).
// Uses V_WMMA_F32_16X16X4_F32 (fp32 in/out -> exact vs fp32 reference).
// =====================================================================
__global__ __launch_bounds__(32) void hg_wmma_gemm(
    const float* __restrict__ Am, long long sA, int lda,
    const float* __restrict__ Bm, long long sB, int ldb,
    float* __restrict__ Cm, long long sC, int ldc,
    const float* __restrict__ bias,
    const float* __restrict__ addsrc, long long sAdd,
    int Kdim, int act)
{
  const int lane = threadIdx.x & 31;
  const int half = lane >> 4;      // K-pair / row-half selector
  const int l    = lane & 15;
  const int m0   = blockIdx.y << 4;
  const int n0   = blockIdx.x << 4;
  const int bz   = blockIdx.z;

  const float* A = Am + (size_t)bz * (size_t)sA;
  const float* Bp = Bm + (size_t)bz * (size_t)sB;
  float*       C = Cm + (size_t)bz * (size_t)sC;

  v8f acc = {0.f,0.f,0.f,0.f,0.f,0.f,0.f,0.f};

  // A fragment (16x4 fp32): lanes 0-15 hold K=k+0/k+1, lanes 16-31 K=k+2/k+3, M=lane&15
  const float* arow = A + (size_t)(m0 + l) * (size_t)lda + (half << 1);
  // B fragment (4x16 fp32): rows k+2*half (v0) and k+2*half+1 (v1), col = n0 + (lane&15)
  const float* bcol = Bp + (size_t)(half << 1) * (size_t)ldb + (n0 + l);

  #pragma unroll 4
  for (int k = 0; k < Kdim; k += 4) {
    v2f a; a.x = arow[k];                 a.y = arow[k + 1];
    v2f b; b.x = bcol[(size_t)k * ldb];   b.y = bcol[(size_t)(k + 1) * ldb];
    acc = __builtin_amdgcn_wmma_f32_16x16x4_f32(
        false, a, false, b, (short)0, acc, false, false);
  }

  const float bv = bias ? bias[n0 + l] : 0.0f;
  #pragma unroll
  for (int r = 0; r < 8; ++r) {
    const int row = m0 + (half << 3) + r;     // C layout: lanes0-15 rows 0..7, lanes16-31 rows 8..15
    float v = acc[r] + bv;
    if (act == 1)      v = tanhf(v);
    else if (act == 2) v = 1.0f / (1.0f + expf(-v));
    if (addsrc) v += addsrc[(size_t)bz * (size_t)sAdd + (size_t)row * ldc + (n0 + l)];
    C[(size_t)row * ldc + (n0 + l)] = v;
  }
}

// ---------------- elementwise helpers ----------------

__global__ void hg_init_state(float* hv, float* zv,
                              const float* __restrict__ h0,
                              const float* __restrict__ z0, int total) {
  int i = blockIdx.x * blockDim.x + threadIdx.x;
  if (i >= total) return;
  int d = i & (DR_ - 1);
  hv[i] = h0[d];
  zv[i] = z0[d];
}

// copy [rows,64] (ld=64) into slot0 of packed buffer (ld=ldp)
__global__ void hg_pack_copy64(const float* __restrict__ src, float* __restrict__ dst,
                               int ldp, int total) {
  int i = blockIdx.x * blockDim.x + threadIdx.x;
  if (i >= total) return;
  int r = i >> 6, j = i & 63;
  dst[(size_t)r * ldp + j] = src[(size_t)r * 64 + j];
}

// P[bn, 0:80] = [ values*masks (t)  |  h ]
__global__ void hg_concat_xh(const float* __restrict__ values,
                             const float* __restrict__ masks, int t,
                             const float* __restrict__ h,
                             float* __restrict__ P, int ldp) {
  int i = blockIdx.x * blockDim.x + threadIdx.x;
  const int total = B_ * N_ * (DIN_ + DR_);
  if (i >= total) return;
  int j = i % 80, bn = i / 80;
  int b = bn >> 10, n = bn & (N_ - 1);
  float v;
  if (j < DIN_) {
    size_t off = (((size_t)b * T_ + t) * N_ + n) * DIN_ + j;
    v = values[off] * masks[off];
  } else {
    v = h[(size_t)bn * DR_ + (j - DIN_)];
  }
  P[(size_t)bn * ldp + j] = v;
}

// P[bn, 0:80] = [ values*masks (t)  |  r*h ]
__global__ void hg_concat_xrh(const float* __restrict__ values,
                              const float* __restrict__ masks, int t,
                              const float* __restrict__ h,
                              const float* __restrict__ r,
                              float* __restrict__ P, int ldp) {
  int i = blockIdx.x * blockDim.x + threadIdx.x;
  const int total = B_ * N_ * (DIN_ + DR_);
  if (i >= total) return;
  int j = i % 80, bn = i / 80;
  int b = bn >> 10, n = bn & (N_ - 1);
  float v;
  if (j < DIN_) {
    size_t off = (((size_t)b * T_ + t) * N_ + n) * DIN_ + j;
    v = values[off] * masks[off];
  } else {
    size_t o = (size_t)bn * DR_ + (j - DIN_);
    v = r[o] * h[o];
  }
  P[(size_t)bn * ldp + j] = v;
}

// GRU blend + obs-select + state update + writes hv_pre / hv_pre*gate / hv_post*gate
__global__ void hg_combine(const float* __restrict__ masks, int t,
                           const float* __restrict__ hv1, const float* __restrict__ uv,
                           const float* __restrict__ cv,
                           const float* __restrict__ zv1, const float* __restrict__ uz,
                           const float* __restrict__ cz,
                           const float* __restrict__ gate,
                           float* __restrict__ hv, float* __restrict__ zv,
                           float* __restrict__ out_pre, float* __restrict__ out_pre_g,
                           float* __restrict__ out_post_g) {
  int bn = blockIdx.x * blockDim.x + threadIdx.x;
  if (bn >= B_ * N_) return;
  int b = bn >> 10, n = bn & (N_ - 1);
  size_t moff = (((size_t)b * T_ + t) * N_ + n) * DIN_;
  float s = 0.f;
  #pragma unroll
  for (int j = 0; j < DIN_; ++j) s += fabsf(masks[moff + j]);
  const float o = (s > 1e-4f) ? 1.f : 0.f;
  size_t base  = (size_t)bn * DR_;
  size_t obase = (((size_t)b * T_ + t) * N_ + n) * DR_;
  for (int d = 0; d < DR_; ++d) {
    float h1 = hv1[base + d], u = uv[base + d], c = cv[base + d];
    float h2 = u * h1 + (1.f - u) * c;
    float hn = h1 + o * (h2 - h1);
    float z1 = zv1[base + d], u2 = uz[base + d], c2 = cz[base + d];
    float z2 = u2 * z1 + (1.f - u2) * c2;
    float zn = z1 + o * (z2 - z1);
    hv[base + d] = hn;
    zv[base + d] = zn;
    float g = gate[base + d];
    out_pre[obase + d]    = h1;
    out_pre_g[obase + d]  = h1 * g;
    out_post_g[obase + d] = hn * g;
  }
}

// =====================================================================
// host orchestration
// =====================================================================
static inline void hg_gemm(hipStream_t s,
    const float* A, long long sA, int lda,
    const float* Bm, long long sB, int ldb,
    float* C, long long sC, int ldc,
    int M, int Ncols, int Kdim,
    const float* bias, const float* add, long long sAdd,
    int act, int batch) {
  dim3 g(Ncols / 16, M / 16, batch), b(32);
  hg_wmma_gemm<<<g, b, 0, s>>>(A, sA, lda, Bm, sB, ldb, C, sC, ldc,
                               bias, add, sAdd, Kdim, act);
}

// one 3-layer diffusion-ODE Euler step: hout = hin + dcode(hin)
static void hg_ode(hipStream_t s, const float* hin, float* hout,
                   const float* A, const float* A2, float* P, float* yt,
                   const float* Win, const float* bin,
                   const float* Wh,  const float* bh,
                   const float* Wout, const float* bout) {
  const int ldp = 3 * DR_;                 // 192
  const long long sP = (long long)N_ * ldp;
  const int BN = B_ * N_;
  const int cgrid = (BN * DR_ + 255) / 256;

  const float* src = hin;
  for (int layer = 0; layer < 3; ++layer) {
    hg_pack_copy64<<<cgrid, 256, 0, s>>>(src, P, ldp, BN * DR_);
    hg_gemm(s, A,  0, N_, P, sP, ldp, P + DR_,     sP, ldp, N_, DR_, N_,
            nullptr, nullptr, 0, 0, B_);
    hg_gemm(s, A2, 0, N_, P, sP, ldp, P + 2 * DR_, sP, ldp, N_, DR_, N_,
            nullptr, nullptr, 0, 0, B_);
    if (layer == 0) {
      hg_gemm(s, P, 0, ldp, Win, 0, DR_, yt, 0, DR_, BN, DR_, ldp,
              bin, nullptr, 0, 1, 1);
      src = yt;
    } else if (layer == 1) {
      hg_gemm(s, P, 0, ldp, Wh, 0, DR_, yt, 0, DR_, BN, DR_, ldp,
              bh, nullptr, 0, 1, 1);
      src = yt;
    } else {
      // hout = hin + (P @ Wout + bout)   (DT == 1.0)
      hg_gemm(s, P, 0, ldp, Wout, 0, DR_, hout, 0, DR_, BN, DR_, ldp,
              bout, hin, 0, 0, 1);
    }
  }
}

// one DCGRU cell: produces r (in yt), u, c  (blend done later in hg_combine)
static void hg_gru(hipStream_t s, const float* values, const float* masks, int t,
                   const float* h, float* P, float* rbuf, float* ubuf, float* cbuf,
                   const float* Wr, const float* br,
                   const float* Wu, const float* bu,
                   const float* Wc, const float* bc,
                   const float* A, const float* A2) {
  const int din = DIN_ + DR_;              // 80
  const int ldp = 3 * din;                 // 240
  const long long sP = (long long)N_ * ldp;
  const int BN = B_ * N_;
  const int cgrid = (BN * din + 255) / 256;

  hg_concat_xh<<<cgrid, 256, 0, s>>>(values, masks, t, h, P, ldp);
  hg_gemm(s, A,  0, N_, P, sP, ldp, P + din,     sP, ldp, N_, din, N_,
          nullptr, nullptr, 0, 0, B_);
  hg_gemm(s, A2, 0, N_, P, sP, ldp, P + 2 * din, sP, ldp, N_, din, N_,
          nullptr, nullptr, 0, 0, B_);
  hg_gemm(s, P, 0, ldp, Wr, 0, DR_, rbuf, 0, DR_, BN, DR_, ldp, br, nullptr, 0, 2, 1);
  hg_gemm(s, P, 0, ldp, Wu, 0, DR_, ubuf, 0, DR_, BN, DR_, ldp, bu, nullptr, 0, 2, 1);

  hg_concat_xrh<<<cgrid, 256, 0, s>>>(values, masks, t, h, rbuf, P, ldp);
  hg_gemm(s, A,  0, N_, P, sP, ldp, P + din,     sP, ldp, N_, din, N_,
          nullptr, nullptr, 0, 0, B_);
  hg_gemm(s, A2, 0, N_, P, sP, ldp, P + 2 * din, sP, ldp, N_, din, N_,
          nullptr, nullptr, 0, 0, B_);
  hg_gemm(s, P, 0, ldp, Wc, 0, DR_, cbuf, 0, DR_, BN, DR_, ldp, bc, nullptr, 0, 1, 1);
}

extern "C" void kernel_launch(void* const* d_in, const int* in_sizes, int n_in,
                              void* d_out, int out_size, void* d_ws, size_t ws_size,
                              hipStream_t stream) {
  // ---- inputs (setup_inputs dict order; params flattened as JAX pytree: sorted keys) ----
  const float* values = (const float*)d_in[0];
  const float* masks  = (const float*)d_in[1];
  const float* Amat   = (const float*)d_in[2];
  const float* tvec   = (const float*)d_in[3];
  const float* gg_Wc = (const float*)d_in[4];  const float* gg_Wr = (const float*)d_in[5];
  const float* gg_Wu = (const float*)d_in[6];  const float* gg_bc = (const float*)d_in[7];
  const float* gg_br = (const float*)d_in[8];  const float* gg_bu = (const float*)d_in[9];
  const float* gv_Wc = (const float*)d_in[10]; const float* gv_Wr = (const float*)d_in[11];
  const float* gv_Wu = (const float*)d_in[12]; const float* gv_bc = (const float*)d_in[13];
  const float* gv_br = (const float*)d_in[14]; const float* gv_bu = (const float*)d_in[15];
  const float* h0 = (const float*)d_in[16];
  const float* og_Wh  = (const float*)d_in[17]; const float* og_Win = (const float*)d_in[18];
  const float* og_Wout= (const float*)d_in[19]; const float* og_bh  = (const float*)d_in[20];
  const float* og_bin = (const float*)d_in[21]; const float* og_bout= (const float*)d_in[22];
  const float* ov_Wh  = (const float*)d_in[23]; const float* ov_Win = (const float*)d_in[24];
  const float* ov_Wout= (const float*)d_in[25]; const float* ov_bh  = (const float*)d_in[26];
  const float* ov_bin = (const float*)d_in[27]; const float* ov_bout= (const float*)d_in[28];
  const float* outW = (const float*)d_in[29]; const float* outb = (const float*)d_in[30];
  const float* z0   = (const float*)d_in[31];
  const float* zW1 = (const float*)d_in[32]; const float* zW2 = (const float*)d_in[33];
  const float* zW3 = (const float*)d_in[34]; const float* zb1 = (const float*)d_in[35];
  const float* zb2 = (const float*)d_in[36]; const float* zb3 = (const float*)d_in[37];
  (void)in_sizes; (void)n_in; (void)out_size; (void)ws_size;

  // ---- output layout (concatenated flat) ----
  float* out = (float*)d_out;
  const size_t O_XPRED = 0;                                   // [8,63,1024,16]
  const size_t O_PRE   = O_XPRED + (size_t)B_*(T_-1)*N_*DIN_; // [8,64,1024,64]
  const size_t O_PREG  = O_PRE   + (size_t)B_*T_*N_*DR_;
  const size_t O_POSTG = O_PREG  + (size_t)B_*T_*N_*DR_;
  const size_t O_TRAJ  = O_POSTG + (size_t)B_*T_*N_*DR_;      // [64]

  // ---- workspace layout (floats) ----
  float* ws = (float*)d_ws;
  const size_t SZ_STATE = (size_t)B_ * N_ * DR_;              // 524288
  float* A2  = ws;                                            // 1048576
  float* hv  = A2  + (size_t)N_ * N_;
  float* zv  = hv  + SZ_STATE;
  float* hv1 = zv  + SZ_STATE;
  float* zv1 = hv1 + SZ_STATE;
  float* P   = zv1 + SZ_STATE;                                // B*N*240
  float* yt  = P   + (size_t)B_ * N_ * 240;                   // ODE tmp / GRU r
  float* uv  = yt  + SZ_STATE;
  float* cv  = uv  + SZ_STATE;
  float* uz  = cv  + SZ_STATE;
  float* cz  = uz  + SZ_STATE;
  float* gA  = cz  + SZ_STATE;
  float* gB  = gA  + SZ_STATE;

  const int BN = B_ * N_;

  // init states: hv = h0, zv = z0 (broadcast)
  hg_init_state<<<(BN * DR_ + 255) / 256, 256, 0, stream>>>(hv, zv, h0, z0, BN * DR_);

  // A2 = A @ A   (fp32 WMMA GEMM, 1024^3)
  hg_gemm(stream, Amat, 0, N_, Amat, 0, N_, A2, 0, N_, N_, N_, N_,
          nullptr, nullptr, 0, 0, 1);

  for (int t = 0; t < T_; ++t) {
    // Euler ODE predictions
    hg_ode(stream, hv, hv1, Amat, A2, P, yt, ov_Win, ov_bin, ov_Wh, ov_bh, ov_Wout, ov_bout);
    hg_ode(stream, zv, zv1, Amat, A2, P, yt, og_Win, og_bin, og_Wh, og_bh, og_Wout, og_bout);

    // DCGRU corrections (r in yt, u/c in dedicated buffers)
    hg_gru(stream, values, masks, t, hv1, P, yt, uv, cv,
           gv_Wr, gv_br, gv_Wu, gv_bu, gv_Wc, gv_bc, Amat, A2);
    hg_gru(stream, values, masks, t, zv1, P, yt, uz, cz,
           gg_Wr, gg_br, gg_Wu, gg_bu, gg_Wc, gg_bc, Amat, A2);

    // gate MLP on zv_pre:  gA=tanh(zv1 W1+b1); gB=tanh(gA W2+b2); gA=sigmoid(gB W3+b3)
    hg_gemm(stream, zv1, 0, DR_, zW1, 0, DR_, gA, 0, DR_, BN, DR_, DR_, zb1, nullptr, 0, 1, 1);
    hg_gemm(stream, gA,  0, DR_, zW2, 0, DR_, gB, 0, DR_, BN, DR_, DR_, zb2, nullptr, 0, 1, 1);
    hg_gemm(stream, gB,  0, DR_, zW3, 0, DR_, gA, 0, DR_, BN, DR_, DR_, zb3, nullptr, 0, 2, 1);

    // blend, update state, emit hv_pre / hv_pre*gate / hv_post*gate
    hg_combine<<<(BN + 255) / 256, 256, 0, stream>>>(
        masks, t, hv1, uv, cv, zv1, uz, cz, gA, hv, zv,
        out + O_PRE, out + O_PREG, out + O_POSTG);

    // X_pred[:, t-1] = (hv_pre * gate)[:, t] @ W + b   (reads the slice just written)
    if (t >= 1) {
      hg_gemm(stream,
              out + O_PREG + (size_t)t * N_ * DR_, (long long)T_ * N_ * DR_, DR_,
              outW, 0, DIN_,
              out + O_XPRED + (size_t)(t - 1) * N_ * DIN_, (long long)(T_-1) * N_ * DIN_, DIN_,
              N_, DIN_, DR_, outb, nullptr, 0, 0, B_);
    }
  }

  // traj_t = t (passthrough)
  hipMemcpyAsync(out + O_TRAJ, tvec, T_ * sizeof(float),
                 hipMemcpyDeviceToDevice, stream);
}